// d_LSKblock_1425929142555
// MI455X (gfx1250) — compile-verified
//
#include <hip/hip_runtime.h>
#include <hip/hip_bf16.h>
#include <math.h>

typedef __attribute__((ext_vector_type(2))) float v2f;
typedef __attribute__((ext_vector_type(4))) float v4f;
typedef __attribute__((ext_vector_type(8))) float v8f;

#define BATCH 4
#define CHN   64
#define HT    56
#define WD    56
#define HW    (HT * WD)           // 3136 (multiple of 16 -> M tiles never cross batch)
#define MTOT  (BATCH * HW)        // 12544
#define MTILES (MTOT / 16)        // 784
#define KRED  288                 // 32 in-channels * 9 taps
#define LDSTR 290                 // padded row stride (floats): even (b64 align)
                                  // and 290*m mod 64 distinct for m=0..15 (banks)

// -------------------------------------------------------------------------
// Grouped 3x3 conv (feature_group_count=2) as LDS-staged implicit GEMM via
// V_WMMA_F32_16X16X4_F32.
//   block  = one (mTile, group); blockDim = nNT waves (32*nNT threads)
//   wave w = one 16-wide N tile (output channels nTile*16..+15)
//   LDS    = A tile [16 x 288] (im2col, padded) + per-wave B tile [16 x 288]
// Inner loop: ds_load_2addr_b64 pairs + wmma, unroll 8.
// -------------------------------------------------------------------------
__global__ void conv3x3_g2_wmma(const float* __restrict__ in,    // (B, 64, H, W)
                                const float* __restrict__ wgt,   // (2*CoutPG, 32, 3, 3)
                                const float* __restrict__ bias,  // (2*CoutPG)
                                float* __restrict__ out,         // (B, 2*CoutPG, H, W)
                                int CoutPG, int nNT)
{
    extern __shared__ float lds[];   // [ (1 + nNT) * 16 * LDSTR ]

    const int tid   = threadIdx.x;
    const int wave  = tid >> 5;
    const int lane  = tid & 31;
    const int blk   = blockIdx.x;
    const int mTile = blk % MTILES;
    const int g     = blk / MTILES;          // group 0/1
    const int mBase = mTile * 16;

    // Tile-constant batch / spatial base: HW % 16 == 0 so no batch crossing.
    const int bb     = mBase / HW;           // batch index for whole tile
    const int hwBase = mBase % HW;           // spatial base for whole tile

    // ---------------- stage A: im2col tile [16 m][288 k] -> lds ----------
    {
        const int tpm = nNT << 1;            // threads per m row (8 or 4)
        const int mS  = tid / tpm;           // 0..15
        const int sub = tid - mS * tpm;
        const int hw  = hwBase + mS;
        const int h   = hw / WD;
        const int wq  = hw % WD;
        const float* inB = in + ((size_t)bb * CHN + (size_t)g * 32) * HW;
        float* dst = &lds[mS * LDSTR];
        for (int ci = sub; ci < 32; ci += tpm) {
            const float* src = inB + (size_t)ci * HW;
#pragma unroll
            for (int r = 0; r < 9; ++r) {
                const int kh = r / 3, kw = r % 3;     // constants (unrolled)
                const int hy = h + kh - 1;
                const int wx = wq + kw - 1;
                float v = 0.0f;
                if ((unsigned)hy < (unsigned)HT && (unsigned)wx < (unsigned)WD)
                    v = src[hy * WD + wx];
                dst[ci * 9 + r] = v;
            }
        }
    }

    // ---------------- stage B: weights [16 n][288 k] per wave -> lds -----
    {
        float* bdst = &lds[(size_t)(1 + wave) * 16 * LDSTR];
        const int nTile = wave;
        __builtin_prefetch(wgt + (size_t)(g * CoutPG) * KRED, 0, 1);
        for (int n = 0; n < 16; ++n) {
            const int ch = nTile * 16 + n;
            const bool v = (ch < CoutPG);
            const float* srcw = wgt + (size_t)(g * CoutPG + (v ? ch : 0)) * KRED;
#pragma unroll
            for (int j = 0; j < 9; ++j) {
                const int k = lane * 9 + j;           // 32 lanes * 9 = 288
                bdst[n * LDSTR + k] = v ? srcw[k] : 0.0f;
            }
        }
    }

    __syncthreads();

    // ---------------- WMMA main loop -------------------------------------
    // Lane mapping (ISA 32-bit 16x4 A / 4x16 B layouts): low half-lanes take
    // K=k0,k0+1, high half-lanes K=k0+2,k0+3; `half` is M row (A) / N col (B).
    const int half = lane & 15;
    const int kOff = (lane < 16) ? 0 : 2;

    const float* Aw = &lds[half * LDSTR + kOff];
    const float* Bw = &lds[(size_t)(1 + wave) * 16 * LDSTR + half * LDSTR + kOff];

    v8f acc = {};
#pragma unroll 8
    for (int k0 = 0; k0 < KRED; k0 += 4) {
        v2f afrag = *(const v2f*)(Aw + k0);   // ds_load_b64 (8B aligned)
        v2f bfrag = *(const v2f*)(Bw + k0);   // ds_load_b64 (8B aligned)
        acc = __builtin_amdgcn_wmma_f32_16x16x4_f32(
            /*neg_a=*/false, afrag, /*neg_b=*/false, bfrag,
            /*c_mod=*/(short)0, acc, /*reuse_a=*/false, /*reuse_b=*/false);
    }

    // ---------------- write-back -----------------------------------------
    // lanes 0-15: N=half, rows M=0..7 ; lanes 16-31: N=half, rows M=8..15.
    // The 8 rows are 8 consecutive floats -> two b128 stores (32B aligned:
    // hwBase multiple of 16, mRow0 in {0,8}).
    const int nOut = wave * 16 + half;
    if (nOut < CoutPG) {
        const int cg    = g * CoutPG + nOut;
        const float bv  = bias[cg];
        const int mRow0 = (lane < 16) ? 0 : 8;
        const int CoutT = 2 * CoutPG;
        float* dst = out + ((size_t)bb * CoutT + cg) * HW + hwBase + mRow0;
        v4f lo = { acc[0] + bv, acc[1] + bv, acc[2] + bv, acc[3] + bv };
        v4f hi = { acc[4] + bv, acc[5] + bv, acc[6] + bv, acc[7] + bv };
        *(v4f*)(dst)     = lo;   // global_store_b128
        *(v4f*)(dst + 4) = hi;   // global_store_b128
    }
}

// -------------------------------------------------------------------------
// Deformable depthwise conv: per output (b,c,h,w), K=k*k taps, bilinear
// sample at (base + learned offset), weighted by depthwise kernel.
// -------------------------------------------------------------------------
__device__ __forceinline__ float dsample(const float* __restrict__ p,
                                         int y, int x) {
    if (y < 0 || y >= HT || x < 0 || x >= WD) return 0.0f;
    return p[y * WD + x];
}

__global__ __launch_bounds__(256)
void deform_dw_kernel(const float* __restrict__ x,       // (B, 64, H, W)
                      const float* __restrict__ offset,  // (B, K*2, H, W)
                      const float* __restrict__ wdw,     // (64, K)
                      const float* __restrict__ bias,    // (64)
                      float* __restrict__ out,           // (B, 64, H, W)
                      int ksz, int pad, int dil)
{
    int idx = blockIdx.x * blockDim.x + threadIdx.x;
    const int total = BATCH * CHN * HW;
    if (idx >= total) return;

    int wq = idx % WD;
    int h  = (idx / WD) % HT;
    int c  = (idx / HW) % CHN;
    int b  = idx / (CHN * HW);
    int K  = ksz * ksz;

    const float* xb   = x + ((size_t)b * CHN + c) * HW;
    const float* offb = offset + (size_t)b * (K * 2) * HW + h * WD + wq;
    const float* wc   = wdw + (size_t)c * K;

    float acc = 0.0f;
    for (int kk = 0; kk < K; ++kk) {
        int ky = kk / ksz;
        int kx = kk - ky * ksz;
        float oy = offb[(size_t)(kk * 2 + 0) * HW];
        float ox = offb[(size_t)(kk * 2 + 1) * HW];
        float py = (float)(h - pad + dil * ky) + oy;
        float px = (float)(wq - pad + dil * kx) + ox;

        float y0f = floorf(py), x0f = floorf(px);
        float ly = py - y0f, lx = px - x0f;
        float hy = 1.0f - ly, hx = 1.0f - lx;
        int y0 = (int)y0f, x0 = (int)x0f;

        float v00 = dsample(xb, y0,     x0);
        float v01 = dsample(xb, y0,     x0 + 1);
        float v10 = dsample(xb, y0 + 1, x0);
        float v11 = dsample(xb, y0 + 1, x0 + 1);
        float val = v00 * hy * hx + v01 * hy * lx
                  + v10 * ly * hx + v11 * ly * lx;
        acc = fmaf(val, wc[kk], acc);
    }
    out[idx] = acc + bias[c];
}

// -------------------------------------------------------------------------
// Global average pool over H*W for both branches: one block per (b,c)
// -------------------------------------------------------------------------
__global__ __launch_bounds__(256)
void gap_kernel(const float* __restrict__ o1, const float* __restrict__ o2,
                float* __restrict__ g1, float* __restrict__ g2)
{
    __shared__ float s1[256];
    __shared__ float s2[256];
    int bc = blockIdx.x;                        // 0 .. B*C-1
    const float* p1 = o1 + (size_t)bc * HW;
    const float* p2 = o2 + (size_t)bc * HW;
    float a1 = 0.0f, a2 = 0.0f;
    for (int i = threadIdx.x; i < HW; i += 256) { a1 += p1[i]; a2 += p2[i]; }
    s1[threadIdx.x] = a1;
    s2[threadIdx.x] = a2;
    __syncthreads();
    for (int s = 128; s > 0; s >>= 1) {
        if ((int)threadIdx.x < s) {
            s1[threadIdx.x] += s1[threadIdx.x + s];
            s2[threadIdx.x] += s2[threadIdx.x + s];
        }
        __syncthreads();
    }
    if (threadIdx.x == 0) {
        g1[bc] = s1[0] * (1.0f / (float)HW);
        g2[bc] = s2[0] * (1.0f / (float)HW);
    }
}

// -------------------------------------------------------------------------
// Softmax over the 2-branch stack axis + weighted sum
// -------------------------------------------------------------------------
__global__ __launch_bounds__(256)
void combine_kernel(const float* __restrict__ o1, const float* __restrict__ o2,
                    const float* __restrict__ g1, const float* __restrict__ g2,
                    float* __restrict__ out)
{
    int idx = blockIdx.x * blockDim.x + threadIdx.x;
    const int total = BATCH * CHN * HW;
    if (idx >= total) return;
    int bc = idx / HW;
    float a = g1[bc], b = g2[bc];
    float mx = fmaxf(a, b);
    float e1 = expf(a - mx);
    float e2 = expf(b - mx);
    float inv = 1.0f / (e1 + e2);
    out[idx] = (e1 * o1[idx] + e2 * o2[idx]) * inv;
}

// -------------------------------------------------------------------------
extern "C" void kernel_launch(void* const* d_in, const int* in_sizes, int n_in,
                              void* d_out, int out_size, void* d_ws, size_t ws_size,
                              hipStream_t stream) {
    const float* x      = (const float*)d_in[0];
    const float* w_off3 = (const float*)d_in[1];
    const float* b_off3 = (const float*)d_in[2];
    const float* w_off2 = (const float*)d_in[3];
    const float* b_off2 = (const float*)d_in[4];
    const float* w7     = (const float*)d_in[5];
    const float* b7     = (const float*)d_in[6];
    const float* w5     = (const float*)d_in[7];
    const float* b5     = (const float*)d_in[8];

    float* ws      = (float*)d_ws;
    float* offset3 = ws;                                   // 4*98*3136 = 1229312
    float* out1    = offset3 + (size_t)BATCH * 98 * HW;    // 802816
    float* offset2 = out1    + (size_t)BATCH * CHN * HW;   // 4*50*3136 = 627200
    float* out2    = offset2 + (size_t)BATCH * 50 * HW;    // 802816
    float* gap1    = out2    + (size_t)BATCH * CHN * HW;   // 256
    float* gap2    = gap1 + BATCH * CHN;                   // 256

    const int totalBC  = BATCH * CHN;                      // 256
    const int totalEl  = BATCH * CHN * HW;                 // 802816
    const int elBlocks = (totalEl + 255) / 256;            // 3136

    const size_t tileB = (size_t)16 * LDSTR * sizeof(float);   // one 16x288 tile

    // offset3 = grouped conv3x3(x): CoutPG=49 -> 4 N-tiles (4 waves/block)
    conv3x3_g2_wmma<<<MTILES * 2, 32 * 4, tileB * (1 + 4), stream>>>(
        x, w_off3, b_off3, offset3, 49, 4);

    // out1 = deform_dw(x, offset3, w7, b7, k=7, pad=9, dil=3)
    deform_dw_kernel<<<elBlocks, 256, 0, stream>>>(
        x, offset3, w7, b7, out1, 7, 9, 3);

    // offset2 = grouped conv3x3(out1): CoutPG=25 -> 2 N-tiles (2 waves/block)
    conv3x3_g2_wmma<<<MTILES * 2, 32 * 2, tileB * (1 + 2), stream>>>(
        out1, w_off2, b_off2, offset2, 25, 2);

    // out2 = deform_dw(out1, offset2, w5, b5, k=5, pad=6, dil=3)
    deform_dw_kernel<<<elBlocks, 256, 0, stream>>>(
        out1, offset2, w5, b5, out2, 5, 6, 3);

    // GAP + softmax gate + merge
    gap_kernel<<<totalBC, 256, 0, stream>>>(out1, out2, gap1, gap2);
    combine_kernel<<<elBlocks, 256, 0, stream>>>(
        out1, out2, gap1, gap2, (float*)d_out);
}